// TextClassifier_59493886984576
// MI455X (gfx1250) — compile-verified
//
#include <hip/hip_runtime.h>
#include <hip/hip_bf16.h>

// EmbeddingBag(mean) + Linear(128->4) fused kernel for MI455X (gfx1250).
//
// Roofline: gather of 819200 x 512B emb rows = ~419 MB >> everything else;
// pure bandwidth problem (~18us at 23.3 TB/s, less once the 51MB table sits
// in the 192MB L2 on replay). FLOPs (~0.1 GFLOP) are negligible, so the win
// is: one wave per bag (no atomics / no searchsorted), register-resident
// 128-dim sum (float4 per lane), and the 16x128 @ 128x4 projection done with
// 32 chained V_WMMA_F32_16X16X4_F32 (full f32 precision, matches reference).
//
// v2: B-matrix feed uses a zero-padded 5-row fc_w table in LDS (row 4 = 0),
// so every lane does an unconditional ds_load_b64 (no per-chunk EXEC
// save/restore, no v_mov zero-fill in the WMMA loop).

#define EMB_DIM        128
#define NUM_CLASSES    4
#define BAGS_PER_BLOCK 16
#define LDS_STRIDE     132   // 16x128 pooled tile, padded row stride (floats)
#define FCW_ROWS       (NUM_CLASSES + 1)   // rows 0-3: fc_w, row 4: zeros

typedef __attribute__((ext_vector_type(2))) float v2f;
typedef __attribute__((ext_vector_type(8))) float v8f;

__global__ __launch_bounds__(512, 2)
void embbag_fc_kernel(const int* __restrict__ text,
                      const int* __restrict__ offsets,
                      const float* __restrict__ emb,
                      const float* __restrict__ fcw,
                      const float* __restrict__ fcb,
                      float* __restrict__ out,
                      int total_tokens, int num_bags)
{
    __shared__ float lds_pooled[BAGS_PER_BLOCK * LDS_STRIDE]; // 8448 B
    __shared__ float lds_fcw[FCW_ROWS * EMB_DIM];             // 2560 B

    const int tid  = threadIdx.x;
    const int wave = tid >> 5;       // 16 waves per block, one bag each
    const int lane = tid & 31;

    // Stage fc_w (4x128) into LDS rows 0-3; row 4 is all zeros (B padding).
    if (tid < NUM_CLASSES * EMB_DIM) lds_fcw[tid] = fcw[tid];
    if (tid < EMB_DIM)               lds_fcw[NUM_CLASSES * EMB_DIM + tid] = 0.0f;

    const int bag = blockIdx.x * BAGS_PER_BLOCK + wave;
    int start = 0, end = 0;
    if (bag < num_bags) {
        start = offsets[bag];
        end   = (bag + 1 < num_bags) ? offsets[bag + 1] : total_tokens;
    }

    // --- Token loop: each lane owns 4 of the 128 dims (coalesced 512B/row).
    // Two accumulator sets + unroll 4 to keep several loads in flight.
    float a0 = 0.f, a1 = 0.f, a2 = 0.f, a3 = 0.f;
    float b0 = 0.f, b1 = 0.f, b2 = 0.f, b3 = 0.f;
    const float4* __restrict__ embv = (const float4*)emb; // 32 float4 per row

    int t = start;
    for (; t + 4 <= end; t += 4) {
        const int tok0 = text[t];
        const int tok1 = text[t + 1];
        const int tok2 = text[t + 2];
        const int tok3 = text[t + 3];
        const float4 v0 = embv[(size_t)tok0 * 32 + lane];
        const float4 v1 = embv[(size_t)tok1 * 32 + lane];
        const float4 v2 = embv[(size_t)tok2 * 32 + lane];
        const float4 v3 = embv[(size_t)tok3 * 32 + lane];
        a0 += v0.x; a1 += v0.y; a2 += v0.z; a3 += v0.w;
        b0 += v1.x; b1 += v1.y; b2 += v1.z; b3 += v1.w;
        a0 += v2.x; a1 += v2.y; a2 += v2.z; a3 += v2.w;
        b0 += v3.x; b1 += v3.y; b2 += v3.z; b3 += v3.w;
    }
    for (; t < end; ++t) {
        const int tok = text[t];
        const float4 v = embv[(size_t)tok * 32 + lane];
        a0 += v.x; a1 += v.y; a2 += v.z; a3 += v.w;
    }
    a0 += b0; a1 += b1; a2 += b2; a3 += b3;

    // Mean (empty bag -> 0, matching reference's max(count,1) with zero sum).
    const int   cnt = end - start;
    const float inv = (cnt > 0) ? (1.0f / (float)cnt) : 0.0f;

    // Pooled tile to LDS: row = local bag, padded stride (16B-aligned store).
    float* dst = &lds_pooled[wave * LDS_STRIDE + lane * 4];
    *(float4*)dst = make_float4(a0 * inv, a1 * inv, a2 * inv, a3 * inv);

    __syncthreads();

    // --- Projection: pooled[16x128] @ fc_w.T[128x4] via 32x WMMA_F32_16X16X4_F32.
    // Wave 0 only (wave-uniform branch -> EXEC all-1s inside, as WMMA requires).
    if (wave == 0) {
        const int m  = lane & 15;          // A: row M   / B,D: col N
        const int kb = (lane >> 4) << 1;   // lanes 0-15: K={0,1}; 16-31: K={2,3}

        // B row: lanes n<4 read fc_w[n][*]; lanes n>=4 read the zero row.
        const int brow = (m < NUM_CLASSES) ? m : NUM_CLASSES;
        const float* __restrict__ bbase = &lds_fcw[brow * EMB_DIM + kb];
        const float* __restrict__ abase = &lds_pooled[m * LDS_STRIDE + kb];

        v8f c = {0.f, 0.f, 0.f, 0.f, 0.f, 0.f, 0.f, 0.f};
        #pragma unroll
        for (int k0 = 0; k0 < EMB_DIM; k0 += 4) {
            // A[m][k] = pooled[m][k0+k]  (ds_load_b64, conflict-free via pad)
            v2f a = *(const v2f*)(abase + k0);
            // B[k][n] = fc_w[n][k0+k] (zero row for n>=4), unconditional load
            v2f b = *(const v2f*)(bbase + k0);
            c = __builtin_amdgcn_wmma_f32_16x16x4_f32(
                    false, a, false, b, (short)0, c, false, false);
        }

        // D layout: VGPR j -> M = (lane>=16 ? 8 : 0)+j, N = lane&15.
        if (m < NUM_CLASSES) {
            const float bias  = fcb[m];
            const int   mrow0 = blockIdx.x * BAGS_PER_BLOCK + ((lane >> 4) * 8);
            #pragma unroll
            for (int j = 0; j < 8; ++j) {
                const int row = mrow0 + j;
                if (row < num_bags)
                    out[(size_t)row * NUM_CLASSES + m] = c[j] + bias;
            }
        }
    }
}

extern "C" void kernel_launch(void* const* d_in, const int* in_sizes, int n_in,
                              void* d_out, int out_size, void* d_ws, size_t ws_size,
                              hipStream_t stream)
{
    const int*   text    = (const int*)d_in[0];
    const int*   offsets = (const int*)d_in[1];
    const float* emb     = (const float*)d_in[2];
    const float* fcw     = (const float*)d_in[3];
    const float* fcb     = (const float*)d_in[4];
    float*       out     = (float*)d_out;

    const int total_tokens = in_sizes[0];
    const int num_bags     = in_sizes[1];
    const int blocks       = (num_bags + BAGS_PER_BLOCK - 1) / BAGS_PER_BLOCK;

    embbag_fc_kernel<<<dim3(blocks), dim3(512), 0, stream>>>(
        text, offsets, emb, fcw, fcb, out, total_tokens, num_bags);
}